// GIN_50268297232946
// MI455X (gfx1250) — compile-verified
//
#include <hip/hip_runtime.h>
#include <math.h>

typedef __attribute__((ext_vector_type(2))) float v2f;
typedef __attribute__((ext_vector_type(8))) float v8f;

#define BN_EPS_F 1e-5f

// ---------------------------------------------------------------------------
// WMMA helper: D = A(16x4 f32) * B(4x16 f32) + C(16x16 f32), wave32.
// ---------------------------------------------------------------------------
__device__ __forceinline__ v8f wmma_f32_16x16x4(v2f a, v2f b, v8f c) {
#if defined(__HIP_DEVICE_COMPILE__)
#if __has_builtin(__builtin_amdgcn_wmma_f32_16x16x4_f32)
  return __builtin_amdgcn_wmma_f32_16x16x4_f32(
      /*neg_a=*/false, a, /*neg_b=*/false, b,
      /*c_mod=*/(short)0, c, /*reuse_a=*/false, /*reuse_b=*/false);
#else
#error "__builtin_amdgcn_wmma_f32_16x16x4_f32 not available on this toolchain"
#endif
#else
  (void)a; (void)b;
  return c;
#endif
}

// ---------------------------------------------------------------------------
// Utility kernels
// ---------------------------------------------------------------------------
__global__ void gin_zero4(float4* __restrict__ p, size_t n4) {
  size_t i = (size_t)blockIdx.x * blockDim.x + threadIdx.x;
  size_t st = (size_t)gridDim.x * blockDim.x;
  float4 z = make_float4(0.f, 0.f, 0.f, 0.f);
  for (; i < n4; i += st) p[i] = z;
}

__global__ void gin_copy4(const float4* __restrict__ s, float4* __restrict__ d, size_t n4) {
  size_t i = (size_t)blockIdx.x * blockDim.x + threadIdx.x;
  size_t st = (size_t)gridDim.x * blockDim.x;
  for (; i < n4; i += st) d[i] = s[i];
}

// histogram count (float, via atomics) -- used for in-degree and graph sizes
__global__ void gin_count(const int* __restrict__ idx, float* __restrict__ cnt, int n) {
  int i = blockIdx.x * blockDim.x + threadIdx.x;
  if (i < n) atomicAdd(&cnt[idx[i]], 1.0f);
}

// p[i] = 1 / max(p[i], 1)
__global__ void gin_recip(float* __restrict__ p, int n) {
  int i = blockIdx.x * blockDim.x + threadIdx.x;
  if (i < n) p[i] = 1.0f / fmaxf(p[i], 1.0f);
}

// ---------------------------------------------------------------------------
// Edge aggregation: agg[dst] += h[src]   (16 threads per edge, float4 rows)
// ---------------------------------------------------------------------------
__global__ void gin_aggregate(const float* __restrict__ h,
                              const int* __restrict__ src,
                              const int* __restrict__ dst,
                              float* __restrict__ agg, int E) {
  long long t = (long long)blockIdx.x * blockDim.x + threadIdx.x;
  int e = (int)(t >> 4);
  if (e >= E) return;
  int c = ((int)t & 15) << 2;
  const float4 v = *(const float4*)(h + (size_t)src[e] * 64 + c);
  float* o = agg + (size_t)dst[e] * 64 + c;
  atomicAdd(o + 0, v.x);
  atomicAdd(o + 1, v.y);
  atomicAdd(o + 2, v.z);
  atomicAdd(o + 3, v.w);
}

// ---------------------------------------------------------------------------
// Fused: u = h + inv_deg*agg ; z = u @ W1 + b1 ; stats += (sum, sumsq) per col
// Block = 128 threads = 4 waves; one 16-row tile; wave w -> cols [16w,16w+16).
// NOTE: z may alias agg (agg is fully staged to LDS before z is written).
// ---------------------------------------------------------------------------
__global__ __launch_bounds__(128) void gin_gemm1_bnstats(
    const float* __restrict__ h, const float* agg,
    const float* __restrict__ inv_deg, const float* __restrict__ W1l,
    const float* __restrict__ b1l, float* z, float* __restrict__ stats) {
  __shared__ float u[16][68];
  const int tid = threadIdx.x;
  const int wave = tid >> 5, lane = tid & 31;
  const int r0 = blockIdx.x << 4;

  {  // stage input tile: 128 threads x 8 floats = 16x64
    const int row = tid >> 3;
    const int c0 = (tid & 7) << 3;
    const size_t base = (size_t)(r0 + row) * 64 + c0;
    const float id = inv_deg[r0 + row];
    float4 h0 = *(const float4*)(h + base);
    float4 h1 = *(const float4*)(h + base + 4);
    float4 a0 = *(const float4*)(agg + base);
    float4 a1 = *(const float4*)(agg + base + 4);
    u[row][c0 + 0] = fmaf(id, a0.x, h0.x);
    u[row][c0 + 1] = fmaf(id, a0.y, h0.y);
    u[row][c0 + 2] = fmaf(id, a0.z, h0.z);
    u[row][c0 + 3] = fmaf(id, a0.w, h0.w);
    u[row][c0 + 4] = fmaf(id, a1.x, h1.x);
    u[row][c0 + 5] = fmaf(id, a1.y, h1.y);
    u[row][c0 + 6] = fmaf(id, a1.z, h1.z);
    u[row][c0 + 7] = fmaf(id, a1.w, h1.w);
  }
  __syncthreads();

  const int nlane = lane & 15;
  const int col = (wave << 4) + nlane;
  const int khalf = (lane >> 4) << 1;  // 0 for lanes 0-15, 2 for lanes 16-31
  v8f acc = {};
#pragma unroll
  for (int k = 0; k < 64; k += 4) {
    v2f a = *(const v2f*)&u[nlane][k + khalf];  // A: M=lane&15, K=khalf..khalf+1
    v2f b;
    b.x = W1l[(size_t)(k + khalf) * 64 + col];  // B: K rows, N=col
    b.y = W1l[(size_t)(k + khalf + 1) * 64 + col];
    acc = wmma_f32_16x16x4(a, b, acc);
  }

  // epilogue: +bias, store z, reduce per-column sum/sumsq for BN
  const float bias = b1l[col];
  const int rbase = r0 + ((lane >> 4) << 3);  // lanes>=16 hold rows +8
  float ps = 0.f, pq = 0.f;
#pragma unroll
  for (int j = 0; j < 8; ++j) {
    float v = acc[j] + bias;
    z[(size_t)(rbase + j) * 64 + col] = v;
    ps += v;
    pq += v * v;
  }
  ps += __shfl_xor(ps, 16, 32);  // merge lane halves (same col, rows +8)
  pq += __shfl_xor(pq, 16, 32);
  if (lane < 16) {
    atomicAdd(&stats[col], ps);
    atomicAdd(&stats[64 + col], pq);
  }
}

// ---------------------------------------------------------------------------
// BN finalize: scale = gamma*rsqrt(var+eps); shift = beta - mean*scale.
// Also re-zeroes stats for the next layer. Launch with 64 threads.
// ---------------------------------------------------------------------------
__global__ void gin_bnfinalize(float* __restrict__ stats, const float* __restrict__ g,
                               const float* __restrict__ b, float* __restrict__ scsh,
                               float invN) {
  int c = threadIdx.x;
  float mean = stats[c] * invN;
  float var = fmaf(-mean, mean, stats[64 + c] * invN);
  float is = rsqrtf(fmaxf(var, 0.f) + BN_EPS_F);
  float sc = g[c] * is;
  scsh[c] = sc;
  scsh[64 + c] = fmaf(-mean, sc, b[c]);
  stats[c] = 0.f;
  stats[64 + c] = 0.f;
}

// ---------------------------------------------------------------------------
// Fused: u = relu(z*scale+shift) ; h_out = u @ W2 + b2 ;
//        node_pool += h_out ; g_pool[batch] += h_out * inv_gcnt
// ---------------------------------------------------------------------------
__global__ __launch_bounds__(128) void gin_gemm2_pool(
    const float* __restrict__ z, const float* __restrict__ scsh,
    const float* __restrict__ W2l, const float* __restrict__ b2l,
    const int* __restrict__ batch, const float* __restrict__ inv_gcnt,
    float* __restrict__ h_out, float* __restrict__ node_pool,
    float* __restrict__ g_pool) {
  __shared__ float u[16][68];
  const int tid = threadIdx.x;
  const int wave = tid >> 5, lane = tid & 31;
  const int r0 = blockIdx.x << 4;

  {  // stage relu(bn(z)) tile
    const int row = tid >> 3;
    const int c0 = (tid & 7) << 3;
    const size_t base = (size_t)(r0 + row) * 64 + c0;
    float4 z0 = *(const float4*)(z + base);
    float4 z1 = *(const float4*)(z + base + 4);
    float zv[8] = {z0.x, z0.y, z0.z, z0.w, z1.x, z1.y, z1.z, z1.w};
#pragma unroll
    for (int i = 0; i < 8; ++i) {
      float sc = scsh[c0 + i];
      float sh = scsh[64 + c0 + i];
      u[row][c0 + i] = fmaxf(fmaf(zv[i], sc, sh), 0.f);
    }
  }
  __syncthreads();

  const int nlane = lane & 15;
  const int col = (wave << 4) + nlane;
  const int khalf = (lane >> 4) << 1;
  v8f acc = {};
#pragma unroll
  for (int k = 0; k < 64; k += 4) {
    v2f a = *(const v2f*)&u[nlane][k + khalf];
    v2f b;
    b.x = W2l[(size_t)(k + khalf) * 64 + col];
    b.y = W2l[(size_t)(k + khalf + 1) * 64 + col];
    acc = wmma_f32_16x16x4(a, b, acc);
  }

  const float bias = b2l[col];
  const int rbase = r0 + ((lane >> 4) << 3);
  const int g0 = batch[r0];
  const int g15 = batch[r0 + 15];
  float vals[8];
  float ps = 0.f;
#pragma unroll
  for (int j = 0; j < 8; ++j) {
    float v = acc[j] + bias;
    vals[j] = v;
    size_t idx = (size_t)(rbase + j) * 64 + col;
    h_out[idx] = v;
    node_pool[idx] += v;
    ps += v;
  }
  if (g0 == g15) {  // whole tile in one graph (batch is sorted): block-reduced pool
    ps += __shfl_xor(ps, 16, 32);
    if (lane < 16) atomicAdd(&g_pool[(size_t)g0 * 64 + col], ps * inv_gcnt[g0]);
  } else {  // boundary tile: per-row atomics
#pragma unroll
    for (int j = 0; j < 8; ++j) {
      int g = batch[rbase + j];
      atomicAdd(&g_pool[(size_t)g * 64 + col], vals[j] * inv_gcnt[g]);
    }
  }
}

// ---------------------------------------------------------------------------
// Host launcher
// ---------------------------------------------------------------------------
static inline int gin_gblocks(size_t n) {
  size_t b = (n + 255) / 256;
  if (b > 4096) b = 4096;
  if (b < 1) b = 1;
  return (int)b;
}

extern "C" void kernel_launch(void* const* d_in, const int* in_sizes, int n_in,
                              void* d_out, int out_size, void* d_ws, size_t ws_size,
                              hipStream_t stream) {
  const float* x     = (const float*)d_in[0];
  const int*   edge  = (const int*)d_in[1];
  const int*   batch = (const int*)d_in[2];
  const float* W1    = (const float*)d_in[3];
  const float* b1    = (const float*)d_in[4];
  const float* gamma = (const float*)d_in[5];
  const float* beta  = (const float*)d_in[6];
  const float* W2    = (const float*)d_in[7];
  const float* b2    = (const float*)d_in[8];

  const int D = 64, L = 4;
  const int N = in_sizes[0] / D;        // 100000 (multiple of 16)
  const int E = in_sizes[1] / 2;        // 1600000
  const int G = out_size / D - N;       // 128
  const int* src = edge;
  const int* dst = edge + E;

  const size_t nd = (size_t)N * D;
  float* ws    = (float*)d_ws;
  float* h_a   = ws;               // N*D  (ping)
  float* h_b   = h_a + nd;         // N*D  (pong)
  float* aggz  = h_b + nd;         // N*D  (agg, then z in place)
  float* deg   = aggz + nd;        // N    (becomes inv_deg)
  float* gcnt  = deg + N;          // G    (becomes inv_gcnt)
  float* stats = gcnt + G;         // 128  (sum | sumsq)
  float* scsh  = stats + 128;      // 128  (scale | shift)
  (void)ws_size; (void)n_in;

  float* node_pool = (float*)d_out;
  float* g_pool    = node_pool + nd;

  const int TB = 256;

  // re-zero all accumulated state every call (graph-replay determinism)
  gin_zero4<<<gin_gblocks((size_t)out_size / 4), TB, 0, stream>>>((float4*)d_out, (size_t)out_size / 4);
  gin_zero4<<<gin_gblocks((size_t)(N + G + 256) / 4), TB, 0, stream>>>((float4*)deg, (size_t)(N + G + 256) / 4);

  // degrees / graph sizes -> reciprocals
  gin_count<<<(E + TB - 1) / TB, TB, 0, stream>>>(dst, deg, E);
  gin_count<<<(N + TB - 1) / TB, TB, 0, stream>>>(batch, gcnt, N);
  gin_recip<<<(N + TB - 1) / TB, TB, 0, stream>>>(deg, N);
  gin_recip<<<(G + TB - 1) / TB, TB, 0, stream>>>(gcnt, G);

  // h0 = x
  gin_copy4<<<gin_gblocks(nd / 4), TB, 0, stream>>>((const float4*)x, (float4*)h_a, nd / 4);

  const int tiles = N / 16;
  for (int l = 0; l < L; ++l) {
    const float* hin = (l & 1) ? h_b : h_a;
    float* hout      = (l & 1) ? h_a : h_b;

    gin_zero4<<<gin_gblocks(nd / 4), TB, 0, stream>>>((float4*)aggz, nd / 4);

    long long tE = (long long)E * 16;
    gin_aggregate<<<(int)((tE + TB - 1) / TB), TB, 0, stream>>>(hin, src, dst, aggz, E);

    gin_gemm1_bnstats<<<tiles, 128, 0, stream>>>(
        hin, aggz, deg, W1 + (size_t)l * D * D, b1 + (size_t)l * D, aggz, stats);

    gin_bnfinalize<<<1, 64, 0, stream>>>(
        stats, gamma + (size_t)l * D, beta + (size_t)l * D, scsh, 1.0f / (float)N);

    gin_gemm2_pool<<<tiles, 128, 0, stream>>>(
        aggz, scsh, W2 + (size_t)l * D * D, b2 + (size_t)l * D,
        batch, gcnt, hout, node_pool, g_pool);
  }
}